// VQCodebookLayer_44693429682348
// MI455X (gfx1250) — compile-verified
//
#include <hip/hip_runtime.h>

typedef __attribute__((ext_vector_type(2))) float v2f;
typedef __attribute__((ext_vector_type(8))) float v8f;

#define NQ      8192      // B*S
#define DIM     64
#define NCODES  65536
#define KSEL    8
#define NC      64        // codebook chunk (codes per LDS stage)
#define NCHUNKS (NCODES / NC)
#define CB_STR  68        // padded dwords per code row in LDS (16B-aligned, conflict-free)
#define SC_STR  65        // padded dwords per score row in LDS
#define WAVES   4
#define TPB     (WAVES * 32)
#define NEG_INF (-3.0e38f)

// ---- Pass 1: |c|^2 for every code (written to workspace, L2-resident) ----
__global__ __launch_bounds__(256) void vq_c2_kernel(const float* __restrict__ cb,
                                                    float* __restrict__ c2) {
    const int n = blockIdx.x * 256 + threadIdx.x;
    const float4* r = (const float4*)(cb + (size_t)n * DIM);
    float s = 0.f;
    #pragma unroll
    for (int i = 0; i < DIM / 4; ++i) {
        const float4 v = r[i];
        s += v.x * v.x + v.y * v.y + v.z * v.z + v.w * v.w;
    }
    c2[n] = s;
}

// ---- Pass 2: fused streaming GEMM (f32 WMMA) + exact per-query top-8 ----
__global__ __launch_bounds__(TPB) void vq_snap_topk_kernel(
    const float* __restrict__ x, const float* __restrict__ cb,
    const float* __restrict__ c2g,
    float* __restrict__ out, int* __restrict__ outIds)
{
    __shared__ float sCb[2][NC * CB_STR];         // double-buffered codebook chunk
    __shared__ float sSc[WAVES][16 * SC_STR];     // per-wave score tile / merge scratch

    const int tid  = threadIdx.x;
    const int lane = tid & 31;
    const int wave = tid >> 5;
    const int half = lane >> 4;          // 0: K-pair {0,1}, 1: K-pair {2,3}
    const int lq   = lane & 15;          // M row / N col / query slot
    const int qbase = (blockIdx.x * WAVES + wave) * 16;

    // Preload A fragments (16 K-steps of 4) for this wave's 16 queries.
    // A 16x4 f32 layout: lane M = lane%16; vgpr0 K = k0+2*half, vgpr1 K+1.
    v2f afrag[16];
    {
        const float* xrow = x + (qbase + lq) * DIM;
        #pragma unroll
        for (int ks = 0; ks < 16; ++ks) {
            const int k = ks * 4 + half * 2;
            afrag[ks].x = xrow[k];
            afrag[ks].y = xrow[k + 1];
        }
    }

    // Async prefetch of one 64x64 chunk into sCb[pbuf]: 8 x B128 per thread-slot.
    auto issue_prefetch = [&](int chunkBase, int pbuf) {
        #pragma unroll
        for (int i = 0; i < (NC * DIM / 4) / TPB; ++i) {  // 8 per thread
            const int f    = tid + i * TPB;               // float4 index 0..1023
            const int code = f >> 4;
            const int part = f & 15;
            const unsigned long long ga =
                (unsigned long long)(const void*)(cb + (size_t)chunkBase * DIM + (size_t)f * 4);
            const unsigned lds = (unsigned)(unsigned long long)(const void*)
                &sCb[pbuf][code * CB_STR + part * 4];
            asm volatile("global_load_async_to_lds_b128 %0, %1, off"
                         :: "v"(lds), "v"(ga) : "memory");
        }
    };

    // Per-lane exact top-8 (query lq, columns half*32..+31 of every chunk).
    float ts[KSEL]; int ti[KSEL];
    #pragma unroll
    for (int i = 0; i < KSEL; ++i) { ts[i] = NEG_INF; ti[i] = 0; }
    float tmin = NEG_INF;

    float* mySc = &sSc[wave][0];

    issue_prefetch(0, 0);
    int buf = 0;

    for (int c = 0; c < NCHUNKS; ++c) {
        const int base = c * NC;

        // L2-hot |c|^2 for this chunk's 4 column tiles (overlaps async wait).
        float c2v[NC / 16];
        #pragma unroll
        for (int nt = 0; nt < NC / 16; ++nt)
            c2v[nt] = c2g[base + nt * 16 + lq];

        // Prefetch next chunk (wraps at the end; harmless) into the other buffer.
        issue_prefetch((base + NC) & (NCODES - 1), buf ^ 1);
        // Previous 8 async loads (this chunk) complete in order -> wait <= 8.
        asm volatile("s_wait_asynccnt 8" ::: "memory");
        __syncthreads();

        const float* chunk = &sCb[buf][0];

        // WMMA: score tile = x . c^T - 0.5*|c|^2 (bias folded into C init).
        #pragma unroll
        for (int nt = 0; nt < NC / 16; ++nt) {
            const int col  = nt * 16 + lq;            // this lane's N column
            const float bias = -0.5f * c2v[nt];
            v8f acc = { bias, bias, bias, bias, bias, bias, bias, bias };
            #pragma unroll
            for (int ks = 0; ks < 16; ++ks) {
                // B 4x16 f32 layout: lane N = lane%16; vgpr0 K = k0+2*half, vgpr1 K+1.
                const float* bp = &chunk[col * CB_STR + ks * 4 + half * 2];
                v2f b; b.x = bp[0]; b.y = bp[1];
                acc = __builtin_amdgcn_wmma_f32_16x16x4_f32(
                    false, afrag[ks], false, b, (short)0, acc, false, false);
            }
            // C/D layout: vgpr i -> row (i + 8*half), col = lane%16 (+ tile base)
            #pragma unroll
            for (int i = 0; i < 8; ++i)
                mySc[(i + half * 8) * SC_STR + col] = acc[i];
        }

        // Per-lane top-8 scan over this lane's half of the chunk.
        #pragma unroll 4
        for (int j = 0; j < NC / 2; ++j) {
            const int col = half * (NC / 2) + j;
            const float s = mySc[lq * SC_STR + col];
            if (s > tmin) {
                const int idx = base + col;
                bool done = false;
                #pragma unroll
                for (int i = 0; i < KSEL; ++i)
                    if (!done && ts[i] == tmin) { ts[i] = s; ti[i] = idx; done = true; }
                tmin = ts[0];
                #pragma unroll
                for (int i = 1; i < KSEL; ++i) tmin = fminf(tmin, ts[i]);
            }
        }

        __syncthreads();   // all waves done reading buf before it is overwritten
        buf ^= 1;
    }

    // ---- Merge the two lanes' candidates per query via LDS ----
    {
        int* myIds = (int*)mySc;
        #pragma unroll
        for (int i = 0; i < KSEL; ++i) {
            mySc[lq * SC_STR + half * 8 + i]        = ts[i];
            myIds[lq * SC_STR + 32 + half * 8 + i]  = ti[i];
        }
    }
    __syncthreads();

    if (lane < 16) {
        const int gq = qbase + lq;
        const int* myIds = (const int*)mySc;
        float cs[16]; int ci[16];
        #pragma unroll
        for (int i = 0; i < 16; ++i) {
            cs[i] = mySc[lq * SC_STR + i];
            ci[i] = myIds[lq * SC_STR + 32 + i];
        }
        int sel[KSEL];
        #pragma unroll
        for (int k = 0; k < KSEL; ++k) {
            float best = NEG_INF; int bidx = 0x7fffffff; int slot = 0;
            #pragma unroll
            for (int i = 0; i < 16; ++i) {
                const bool better = (cs[i] > best) || (cs[i] == best && ci[i] < bidx);
                if (better) { best = cs[i]; bidx = ci[i]; slot = i; }
            }
            sel[k] = ci[slot];
            cs[slot] = NEG_INF;
            outIds[gq * KSEL + k] = sel[k];
        }
        // Mean of the 8 selected code vectors (L2-resident gathers).
        float4* o = (float4*)(out + (size_t)gq * DIM);
        #pragma unroll
        for (int db = 0; db < DIM / 4; ++db) {
            float4 a = make_float4(0.f, 0.f, 0.f, 0.f);
            #pragma unroll
            for (int k = 0; k < KSEL; ++k) {
                const float4 r = ((const float4*)(cb + (size_t)sel[k] * DIM))[db];
                a.x += r.x; a.y += r.y; a.z += r.z; a.w += r.w;
            }
            a.x *= 0.125f; a.y *= 0.125f; a.z *= 0.125f; a.w *= 0.125f;
            o[db] = a;
        }
    }
}

extern "C" void kernel_launch(void* const* d_in, const int* in_sizes, int n_in,
                              void* d_out, int out_size, void* d_ws, size_t ws_size,
                              hipStream_t stream) {
    (void)in_sizes; (void)n_in; (void)out_size; (void)ws_size;
    const float* x  = (const float*)d_in[0];
    const float* cb = (const float*)d_in[1];
    float* c2  = (float*)d_ws;                       // 65536 floats (256 KB scratch)
    float* out = (float*)d_out;                      // [8192, 64] f32
    int*   ids = (int*)d_out + (size_t)NQ * DIM;     // [8192, 8] int32 follows

    vq_c2_kernel<<<NCODES / 256, 256, 0, stream>>>(cb, c2);
    const int blocks = NQ / (WAVES * 16);            // 128
    vq_snap_topk_kernel<<<blocks, TPB, 0, stream>>>(x, cb, c2, out, ids);
}